// KNNHG_84748294684745
// MI455X (gfx1250) — compile-verified
//
#include <hip/hip_runtime.h>

// Problem constants (from reference: x = (16, 256, 768) fp32, k = 16)
#define BB   16
#define SS   256
#define DD   768
#define NN   4096          // BB*SS
#define KNN  16
#define WAVES_PER_BLK 8    // 256 threads, wave32
#define PF   4             // K-loop pipeline stages (4 K-values each)

typedef __attribute__((ext_vector_type(2))) float v2f;
typedef __attribute__((ext_vector_type(4))) float v4f;
typedef __attribute__((ext_vector_type(8))) float v8f;

// ---------------------------------------------------------------------------
// Kernel 1: sq[i] = ||x_i||^2  (one wave32 per row, b128 coalesced, shfl reduce)
// ---------------------------------------------------------------------------
__global__ __launch_bounds__(256) void knn_row_norms(const float* __restrict__ x,
                                                     float* __restrict__ sq) {
    const int wave = threadIdx.x >> 5;
    const int lane = threadIdx.x & 31;
    const int row  = blockIdx.x * WAVES_PER_BLK + wave;
    const v4f* xr = (const v4f*)(x + (size_t)row * DD);
    float s = 0.0f;
    #pragma unroll
    for (int i = lane; i < DD / 4; i += 32) {   // 6 iterations, 128B/clk per wave
        v4f v = xr[i];
        s += v.x * v.x + v.y * v.y + v.z * v.z + v.w * v.w;
    }
    #pragma unroll
    for (int off = 16; off > 0; off >>= 1)
        s += __shfl_xor(s, off, 32);
    if (lane == 0) sq[row] = s;
}

// Branch-free sorted insertion of (d, idx) into ascending top-16 kept in VGPRs.
__device__ __forceinline__ void insert16(float d, int idx,
                                         float (&bd)[KNN], int (&bi)[KNN]) {
    float nd = d; int ni = idx;
    #pragma unroll
    for (int j = 0; j < KNN; ++j) {
        bool lt = nd < bd[j];
        float sd = lt ? bd[j] : nd;
        int   si = lt ? bi[j] : ni;
        bd[j] = lt ? nd : bd[j];
        bi[j] = lt ? ni : bi[j];
        nd = sd; ni = si;
    }
}

// ---------------------------------------------------------------------------
// Kernel 2: fused Gram-tile (fp32 WMMA, software-pipelined K) + per-row top-16
//           + histogram output. Block b owns rows [16b, 16b+16).
// ---------------------------------------------------------------------------
__global__ __launch_bounds__(256) void knn_hypergraph(const float* __restrict__ x,
                                                      const float* __restrict__ sq,
                                                      float* __restrict__ out) {
    // Stride 20 floats = 80B: 16B-aligned rows AND bank-conflict-free for both
    // the half-wave tile writes and the per-row b128 reads.
    __shared__ __align__(16) float tileLDS[WAVES_PER_BLK][16][20];   // 10 KB
    __shared__ float mergeD[16][WAVES_PER_BLK * KNN];                //  8 KB
    __shared__ int   mergeI[16][WAVES_PER_BLK * KNN];                //  8 KB
    // Output histogram reuses tileLDS (temporally disjoint, barrier-separated).
    float* counts = &tileLDS[0][0][0];                               // 16*256 floats

    const int tid  = threadIdx.x;
    const int wave = tid >> 5;
    const int lane = tid & 31;
    const int l16  = lane & 15;
    const int lhi  = lane >> 4;                // 0 or 1: selects K pair
    const int rowBase = blockIdx.x * 16;

    // A-fragment base: lane holds row (rowBase + l16), K pair starting at 2*lhi
    const float* arow = x + (size_t)(rowBase + l16) * DD + 2 * lhi;

    // Per-lane selection state: lanes 0..15 own local rows 0..15.
    float bd[KNN]; int bi[KNN];
    #pragma unroll
    for (int j = 0; j < KNN; ++j) { bd[j] = __FLT_MAX__; bi[j] = 0x7fffffff; }

    for (int ct = wave; ct < NN / 16; ct += WAVES_PER_BLK) {
        const int colBase = ct * 16;
        const float* brow = x + (size_t)(colBase + l16) * DD + 2 * lhi;

        // Speculative prefetch of this wave's next column tile (L2-resident).
        if (ct + WAVES_PER_BLK < NN / 16)
            __builtin_prefetch(brow + (size_t)(16 * WAVES_PER_BLK) * DD, 0, 0);

        // --- fp32 WMMA with PF-stage software pipeline over K = 768 ---
        v2f abuf[PF], bbuf[PF];
        #pragma unroll
        for (int p = 0; p < PF; ++p) {
            abuf[p] = *(const v2f*)(arow + 4 * p);
            bbuf[p] = *(const v2f*)(brow + 4 * p);
        }
        v8f c = {};
        #pragma unroll 2
        for (int kk = 0; kk < DD - 4 * PF; kk += 4 * PF) {
            v2f anx[PF], bnx[PF];
            #pragma unroll
            for (int p = 0; p < PF; ++p) {     // issue next group's 8 loads first
                anx[p] = *(const v2f*)(arow + kk + 4 * PF + 4 * p);
                bnx[p] = *(const v2f*)(brow + kk + 4 * PF + 4 * p);
            }
            #pragma unroll
            for (int p = 0; p < PF; ++p)       // WMMA on current group meanwhile
                c = __builtin_amdgcn_wmma_f32_16x16x4_f32(
                        false, abuf[p], false, bbuf[p], (short)0, c, false, false);
            #pragma unroll
            for (int p = 0; p < PF; ++p) { abuf[p] = anx[p]; bbuf[p] = bnx[p]; }
        }
        #pragma unroll
        for (int p = 0; p < PF; ++p)           // drain last group
            c = __builtin_amdgcn_wmma_f32_16x16x4_f32(
                    false, abuf[p], false, bbuf[p], (short)0, c, false, false);

        // d(m,n) = sq[col] - 2*dot  (sq[row] is row-constant: irrelevant to ranking)
        const float sqc = sq[colBase + l16];
        #pragma unroll
        for (int v = 0; v < 8; ++v) {
            const int m = v + 8 * lhi;         // C/D layout: M = v + 8*(lane/16)
            tileLDS[wave][m][l16] = __builtin_fmaf(-2.0f, c[v], sqc);
        }
        // Wave-internal LDS producer->consumer ordering (CDNA5 split counters)
        asm volatile("s_wait_dscnt 0" ::: "memory");

        // Lanes 0..15: scan own row as 4x b128, keep running top-16.
        if (lane < 16) {
            const v4f* rowp = (const v4f*)&tileLDS[wave][lane][0];
            #pragma unroll
            for (int q = 0; q < 4; ++q) {
                v4f d4 = rowp[q];
                #pragma unroll
                for (int e = 0; e < 4; ++e) {
                    const float d = d4[e];
                    if (d < bd[KNN - 1])
                        insert16(d, colBase + 4 * q + e, bd, bi);
                }
            }
        }
    }

    // Publish per-wave winners.
    if (lane < 16) {
        #pragma unroll
        for (int j = 0; j < KNN; ++j) {
            mergeD[lane][wave * KNN + j] = bd[j];
            mergeI[lane][wave * KNN + j] = bi[j];
        }
    }
    __syncthreads();   // all waves done with tileLDS; merge arrays complete

    // Zero the histogram (aliases tileLDS).
    for (int i = tid; i < 16 * SS; i += 256)
        counts[i] = 0.0f;
    __syncthreads();

    // Final merge: 16 threads (one per row, spread across all 8 waves).
    if ((tid & 15) == 0) {
        const int row = tid >> 4;
        float fd[KNN]; int fi[KNN];
        #pragma unroll
        for (int j = 0; j < KNN; ++j) { fd[j] = __FLT_MAX__; fi[j] = 0x7fffffff; }
        for (int cnd = 0; cnd < WAVES_PER_BLK * KNN; ++cnd) {
            const float d = mergeD[row][cnd];
            if (d < fd[KNN - 1])
                insert16(d, mergeI[row][cnd], fd, fi);
        }
        // Histogram over within-batch position s' = idx mod 256.
        #pragma unroll
        for (int j = 0; j < KNN; ++j)
            counts[row * SS + (fi[j] & (SS - 1))] += 1.0f;
    }
    __syncthreads();

    // Coalesced store: block owns out rows [rowBase, rowBase+16) of [N, 256].
    float* obase = out + (size_t)rowBase * SS;
    for (int i = tid; i < 16 * SS; i += 256)
        obase[i] = counts[i];
}

// ---------------------------------------------------------------------------
extern "C" void kernel_launch(void* const* d_in, const int* in_sizes, int n_in,
                              void* d_out, int out_size, void* d_ws, size_t ws_size,
                              hipStream_t stream) {
    const float* x   = (const float*)d_in[0];  // (16, 256, 768) fp32
    float*       out = (float*)d_out;          // (16, 256, 256) fp32
    float*       sq  = (float*)d_ws;           // 4096 fp32 row norms

    knn_row_norms<<<NN / WAVES_PER_BLK, 256, 0, stream>>>(x, sq);
    knn_hypergraph<<<NN / 16, 256, 0, stream>>>(x, sq, out);
}